// GaussianMixturePosterior_64269890617656
// MI455X (gfx1250) — compile-verified
//
#include <hip/hip_runtime.h>
#include <math.h>

// Problem constants (from reference)
#define B_DIM 4
#define D_DIM 512
#define N_DIM 1024
#define K_DIM 64
#define EPSV 1e-10f
#define LOG_CONST 470.4965290007924f   // (512/2)*ln(2*pi)

typedef __attribute__((ext_vector_type(2))) float v2f;
typedef __attribute__((ext_vector_type(8))) float v8f;
typedef __attribute__((ext_vector_type(4))) float f4;

#if __has_builtin(__builtin_amdgcn_global_load_async_to_lds_b32)
#define HAVE_ASYNC_LDS 1
#else
#define HAVE_ASYNC_LDS 0
#endif

#if HAVE_ASYNC_LDS
typedef __attribute__((address_space(1))) int as1_int;   // global (HIP "__device__")
typedef __attribute__((address_space(3))) int as3_int;   // LDS

__device__ __forceinline__ void async_g2lds_b32(const float* gsrc, const float* ldst) {
    // Route through integers: as1 = full 64-bit VA, as3 = low 32 bits (LDS offset,
    // per the ISA flat-aperture rule).
    as1_int* g = (as1_int*)(unsigned long long)gsrc;
    as3_int* l = (as3_int*)(unsigned int)(unsigned long long)ldst;
    __builtin_amdgcn_global_load_async_to_lds_b32(g, l, 0, 0);
}

__device__ __forceinline__ void wait_asynccnt0() {
#if __has_builtin(__builtin_amdgcn_s_wait_asynccnt)
    __builtin_amdgcn_s_wait_asynccnt(0);
#else
    asm volatile("s_wait_asynccnt 0x0" ::: "memory");
#endif
}
#endif

// ---------------------------------------------------------------------------
// Kernel K0: build GEMM A-matrix W[k][0:512]=inv_var, W[k][512:1024]=-2*inv_var*mu
// and s0[k] = sum_d inv_var*mu^2.  One block per k, LDS reduction.
// ---------------------------------------------------------------------------
__global__ __launch_bounds__(256)
void prep_kernel(const float* __restrict__ mu, const float* __restrict__ logvar,
                 float* __restrict__ Aw, float* __restrict__ s0) {
    __shared__ float red[256];
    const int k   = blockIdx.x;
    const int tid = threadIdx.x;
    float acc = 0.f;
    for (int d = tid; d < D_DIM; d += 256) {
        float lv  = logvar[k * D_DIM + d];
        float m   = mu[k * D_DIM + d];
        float var = expf(lv) + EPSV;
        float iv  = 1.0f / fmaxf(var, EPSV);
        Aw[k * (2 * D_DIM) + d]          = iv;
        Aw[k * (2 * D_DIM) + D_DIM + d]  = -2.0f * iv * m;
        acc += iv * m * m;
    }
    red[tid] = acc;
    __syncthreads();
    for (int s = 128; s > 0; s >>= 1) {
        if (tid < s) red[tid] += red[tid + s];
        __syncthreads();
    }
    if (tid == 0) s0[k] = red[0];
}

// ---------------------------------------------------------------------------
// Kernel P: mahalanobis via V_WMMA_F32_16X16X4_F32 + fused bias/max/clip/softmax.
// Block = 128 threads (4 waves).  Wave w computes k-tile [16w,16w+16) x n-tile
// of 16 for one b; the 64x16 exponent tile is reduced over K in LDS.
// A-frag (16x4 f32): lanes 0-15 hold {K0,K1} for M=lane, lanes 16-31 {K2,K3}.
// C/D (16x16 f32, v8f): row M = v + 8*(lane>=16), col N = lane%16.
// ---------------------------------------------------------------------------
__global__ __launch_bounds__(128)
void posterior_kernel(const float* __restrict__ x, const float* __restrict__ Aw,
                      const float* __restrict__ s0, const float* __restrict__ bias,
                      const float* __restrict__ alpha_p, float* __restrict__ prob) {
    __shared__ float exl[K_DIM * 17];       // 64 rows x 16 cols, padded stride 17
    const int tid   = threadIdx.x;
    const int wave  = tid >> 5;
    const int lane  = tid & 31;
    const int half  = lane >> 4;            // 0: K0/K1 slice, 1: K2/K3 slice
    const int row16 = lane & 15;
    const int blocksPerB = N_DIM / 16;      // 64
    const int b     = blockIdx.x / blocksPerB;
    const int nBase = (blockIdx.x % blocksPerB) * 16;
    const int kBase = wave * 16;

    const float* xb = x + (size_t)b * D_DIM * N_DIM;
    const int n     = nBase + row16;
    const int krow  = kBase + row16;

    v8f c = {};
    // Phase 1: j in [0,512): A = inv_var, B = x^2
    for (int j = 0; j < D_DIM; j += 4) {
        const int ja = j + 2 * half;
        v2f a, bf;
        a.x = Aw[krow * (2 * D_DIM) + ja];
        a.y = Aw[krow * (2 * D_DIM) + ja + 1];
        float x0 = xb[(size_t)ja * N_DIM + n];
        float x1 = xb[(size_t)(ja + 1) * N_DIM + n];
        bf.x = x0 * x0;
        bf.y = x1 * x1;
        c = __builtin_amdgcn_wmma_f32_16x16x4_f32(false, a, false, bf,
                                                  (short)0, c, false, false);
    }
    // Phase 2: A = -2*inv_var*mu, B = x
    for (int j = 0; j < D_DIM; j += 4) {
        const int ja = j + 2 * half;
        v2f a, bf;
        a.x = Aw[krow * (2 * D_DIM) + D_DIM + ja];
        a.y = Aw[krow * (2 * D_DIM) + D_DIM + ja + 1];
        bf.x = xb[(size_t)ja * N_DIM + n];
        bf.y = xb[(size_t)(ja + 1) * N_DIM + n];
        c = __builtin_amdgcn_wmma_f32_16x16x4_f32(false, a, false, bf,
                                                  (short)0, c, false, false);
    }

    const float alpha = alpha_p[0];
    #pragma unroll
    for (int v = 0; v < 8; ++v) {
        const int k   = kBase + v + 8 * half;
        float mah     = c[v] + s0[k];
        float ex      = -0.5f * alpha * mah - LOG_CONST + bias[k];
        exl[k * 17 + row16] = ex;
    }
    __syncthreads();

    // Softmax over K=64 per column (16 columns, serial 64-loops: trivial work)
    if (tid < 16) {
        const int nn = tid;
        float m = -__builtin_inff();
        for (int k = 0; k < K_DIM; ++k) m = fmaxf(m, exl[k * 17 + nn]);
        float sum = 0.f;
        for (int k = 0; k < K_DIM; ++k) {
            float e = exl[k * 17 + nn] - m;
            e = fminf(fmaxf(e, -100.f), 100.f);
            e = expf(e);
            exl[k * 17 + nn] = e;
            sum += e;
        }
        const float inv = 1.0f / sum;
        for (int k = 0; k < K_DIM; ++k) {
            prob[(size_t)b * K_DIM * N_DIM + (size_t)k * N_DIM + nBase + nn] =
                exl[k * 17 + nn] * inv;
        }
    }
}

// ---------------------------------------------------------------------------
// Kernel D: stream the 537 MB diff tensor (the bandwidth-dominant output).
// Block handles (b, 32-wide n-tile, 8-wide k-tile).  x tile is staged in LDS
// transposed via CDNA5 async global->LDS copies (ASYNCcnt path) when the
// toolchain exposes the builtin, then global writes are fully coalesced
// 512B-per-wave b128 stores along D with a NON-TEMPORAL hint so the 537 MB
// stream doesn't sweep L2 (keeps the 64x-reused x tiles resident).
// ---------------------------------------------------------------------------
#define TN 32
#define TK 8
#define XPAD 516   // 512 + 4: keeps f4 16B alignment, spreads banks

__global__ __launch_bounds__(256)
void diff_kernel(const float* __restrict__ x, const float* __restrict__ mu,
                 float* __restrict__ diff) {
    __shared__ float xl[TN * XPAD];        // ~64.5 KB
    __shared__ float ml[TK * D_DIM];       // 16 KB
    const int tid = threadIdx.x;
    const int blocksPerB = N_DIM / TN;     // 32
    const int b     = blockIdx.x / blocksPerB;
    const int nBase = (blockIdx.x % blocksPerB) * TN;
    const int kBase = blockIdx.y * TK;

    const float* xb = x + (size_t)b * D_DIM * N_DIM;

    // Stage x[b, :, nBase:nBase+32] into LDS, transposed to [n][d].
    // Global side is coalesced (consecutive lanes -> consecutive n).
#if HAVE_ASYNC_LDS
    for (int idx = tid; idx < D_DIM * TN; idx += 256) {
        const int d  = idx >> 5;           // / TN
        const int nn = idx & (TN - 1);
        async_g2lds_b32(&xb[(size_t)d * N_DIM + nBase + nn], &xl[nn * XPAD + d]);
    }
    wait_asynccnt0();                      // this wave's async copies done
#else
    for (int idx = tid; idx < D_DIM * TN; idx += 256) {
        const int d  = idx >> 5;
        const int nn = idx & (TN - 1);
        xl[nn * XPAD + d] = xb[(size_t)d * N_DIM + nBase + nn];
    }
#endif
    // Coalesced load of mu rows
    for (int idx = tid; idx < TK * D_DIM; idx += 256) {
        const int kk = idx >> 9;           // / D_DIM
        const int d  = idx & (D_DIM - 1);
        ml[kk * D_DIM + d] = mu[(size_t)(kBase + kk) * D_DIM + d];
    }
    __syncthreads();                       // all waves' staging visible

    for (int kk = 0; kk < TK; ++kk) {
        const size_t base =
            (((size_t)b * K_DIM + kBase + kk) * N_DIM + nBase) * D_DIM;
        for (int idx = tid; idx < TN * (D_DIM / 4); idx += 256) {
            const int nn = idx >> 7;               // / (D/4)
            const int d4 = (idx & 127) << 2;
            f4 xv = *(const f4*)&xl[nn * XPAD + d4];
            f4 mv = *(const f4*)&ml[kk * D_DIM + d4];
            f4 r  = xv - mv;
            // streaming store: non-temporal b128, fully coalesced along D
            __builtin_nontemporal_store(r, (f4*)&diff[base + (size_t)nn * D_DIM + d4]);
        }
    }
}

// ---------------------------------------------------------------------------
extern "C" void kernel_launch(void* const* d_in, const int* in_sizes, int n_in,
                              void* d_out, int out_size, void* d_ws, size_t ws_size,
                              hipStream_t stream) {
    (void)in_sizes; (void)n_in; (void)out_size; (void)ws_size;
    const float* x      = (const float*)d_in[0];
    const float* mu     = (const float*)d_in[1];
    const float* logvar = (const float*)d_in[2];
    const float* bias   = (const float*)d_in[3];
    const float* alpha  = (const float*)d_in[4];

    float* out  = (float*)d_out;
    float* prob = out;                                        // (B,K,N)
    float* diff = out + (size_t)B_DIM * K_DIM * N_DIM;        // (B,K,N,D)

    float* Aw = (float*)d_ws;                 // 64 x 1024 f32 = 256 KB
    float* s0 = Aw + K_DIM * 2 * D_DIM;       // 64 f32

    prep_kernel<<<K_DIM, 256, 0, stream>>>(mu, logvar, Aw, s0);
    posterior_kernel<<<B_DIM * (N_DIM / 16), 128, 0, stream>>>(
        x, Aw, s0, bias, alpha, prob);
    diff_kernel<<<dim3(B_DIM * (N_DIM / TN), K_DIM / TK), 256, 0, stream>>>(
        x, mu, diff);
}